// HashEncoder_67963562492047
// MI455X (gfx1250) — compile-verified
//
#include <hip/hip_runtime.h>
#include <math.h>

#define NUM_LEVELS 16
#define HASH_SIZE  (1u << 19)           // 2^19, divides 2^32 -> uint32 hash math is exact

typedef __attribute__((ext_vector_type(2))) float v2f;

struct ResArr { float r[NUM_LEVELS]; };

// One thread per (point, level) pair.
//   pair index i : n = i >> 4, l = i & 15
// Wave32 layout: lanes 0-15 -> point n (levels 0..15), lanes 16-31 -> point n+1.
// => each wave's float2 store instruction covers 256 contiguous bytes (perfect coalescing).
// Embedding gathers use default (RT) temporal policy so the 64 MB of tables stays in the
// 192 MB L2; the 256 MB output stream and 24 MB x stream use non-temporal hints (CDNA5 TH
// field) so they don't evict the tables.
__global__ __launch_bounds__(256) void HashEncoder_kernel(
    const float* __restrict__ x,       // [N,3] f32
    const v2f*  __restrict__ emb,      // [16, 2^19] float2
    v2f*        __restrict__ out,      // [N, 16] float2  (== [N,32] f32)
    ResArr res,
    unsigned npairs)                   // N * 16
{
    __shared__ float s_res[NUM_LEVELS];
    if (threadIdx.x < NUM_LEVELS) s_res[threadIdx.x] = res.r[threadIdx.x];
    __syncthreads();

    const unsigned tid    = blockIdx.x * blockDim.x + threadIdx.x;
    const unsigned stride = gridDim.x * blockDim.x;   // blockDim=256 -> always multiple of 16,
                                                      // so the level of this thread never changes
    const unsigned l       = tid & 15u;
    const float    rl      = s_res[l];                // level resolution (f32, matches reference)
    const unsigned embBase = l * HASH_SIZE;           // float2 index of this level's table

    #pragma unroll 4
    for (unsigned i = tid; i < npairs; i += stride) {
        const unsigned n  = i >> 4;
        const float*   xp = x + n * 3u;
        // streaming read-once coords: non-temporal loads (16 lanes share each address ->
        // single broadcast transaction per cacheline within the wave)
        const float x0 = __builtin_nontemporal_load(xp + 0);
        const float x1 = __builtin_nontemporal_load(xp + 1);
        const float x2 = __builtin_nontemporal_load(xp + 2);

        // x in [0,1), rl <= 2048 -> products in [0, 2048): f32->u32 truncation == floor.
        // Reference hash is ((floor*primes).sum mod 2^19) on non-negative int64;
        // since 2^19 | 2^32 this is exactly reproduced by wrapping uint32 arithmetic.
        const unsigned i0 = (unsigned)(x0 * rl);
        const unsigned i1 = (unsigned)(x1 * rl);
        const unsigned i2 = (unsigned)(x2 * rl);
        const unsigned h  = (i0 + i1 * 2654435761u + i2 * 805459861u) & (HASH_SIZE - 1u);

        // random 8B gather, L2-resident table (regular temporal policy)
        const v2f f = emb[embBase + h];

        // write-once 256 MB stream: non-temporal store keeps L2 for the hash tables
        __builtin_nontemporal_store(f, out + (n << 4) + l);
    }
}

extern "C" void kernel_launch(void* const* d_in, const int* in_sizes, int n_in,
                              void* d_out, int out_size, void* d_ws, size_t ws_size,
                              hipStream_t stream) {
    (void)n_in; (void)out_size; (void)d_ws; (void)ws_size;

    const float* x   = (const float*)d_in[0];   // [N,3] f32
    const v2f*   emb = (const v2f*)d_in[1];     // [16, 2^19, 2] f32
    v2f*         out = (v2f*)d_out;             // [N,32] f32 viewed as [N,16] float2

    const unsigned n_points = (unsigned)(in_sizes[0] / 3);
    const unsigned npairs   = n_points * 16u;

    // Reproduce RESOLUTIONS exactly as Python computes them (double pow + int truncation,
    // then cast to f32). Level 15 lands exactly on the 2048.0 truncation boundary, so we
    // must mirror the host libm computation rather than bake a guess.
    ResArr res;
    const double b = pow(2048.0 / 16.0, 1.0 / 15.0);
    for (int i = 0; i < NUM_LEVELS; ++i)
        res.r[i] = (float)(int)(16.0 * pow(b, (double)i));

    const unsigned block = 256u;
    const unsigned iters = 4u;                  // pairs per thread (matches unroll factor)
    unsigned grid = (npairs + block * iters - 1u) / (block * iters);
    if (grid < 1u) grid = 1u;

    HashEncoder_kernel<<<dim3(grid), dim3(block), 0, stream>>>(x, emb, out, res, npairs);
}